// GCNPool_2860448219409
// MI455X (gfx1250) — compile-verified
//
#include <hip/hip_runtime.h>
#include <hip/hip_bf16.h>

// ---------------- problem constants (from reference) ----------------
#define GNUM   8
#define NNODE  4096
#define SSAMP  2048
#define KNB    64
#define EPER   65536
#define CCLS   13
#define NT     (GNUM * NNODE)     // 32768
#define ETOT   (GNUM * EPER)      // 524288
#define GS     (GNUM * SSAMP)     // 16384
#define R2     (0.4f * 0.4f)

typedef float v2f __attribute__((ext_vector_type(2)));
typedef float v8f __attribute__((ext_vector_type(8)));

// ---------------- WMMA f32 GEMM: Y[M,N] = X[M,K] @ W[K,N] ----------------
// One wave per 16x16 output tile, K consumed in chunks of 4 via
// V_WMMA_F32_16X16X4_F32.  Requires M%16==0, N%16==0, K%4==0.
__global__ __launch_bounds__(128) void gemm_wmma_f32(
    const float* __restrict__ X, const float* __restrict__ W,
    float* __restrict__ Y, int M, int K, int N) {
  int wave = (blockIdx.x * blockDim.x + threadIdx.x) >> 5;
  int lane = threadIdx.x & 31;
  int tilesN = N >> 4;
  int totalTiles = (M >> 4) * tilesN;
  if (wave >= totalTiles) return;                 // wave-uniform exit
  int tm = (wave / tilesN) << 4;
  int tn = (wave % tilesN) << 4;
  int r = lane & 15;                               // A row / B col within tile
  int khalf = (lane >> 4) << 1;                    // 0 (lanes 0-15) or 2 (16-31)
  const float* Arow = X + (size_t)(tm + r) * K;
  const float* Bcol = W + tn + r;
  v8f acc = {};
  for (int kb = 0; kb < K; kb += 4) {
    v2f a, b;
    a.x = Arow[kb + khalf];
    a.y = Arow[kb + khalf + 1];
    b.x = Bcol[(size_t)(kb + khalf) * N];
    b.y = Bcol[(size_t)(kb + khalf + 1) * N];
    acc = __builtin_amdgcn_wmma_f32_16x16x4_f32(
        /*neg_a=*/false, a, /*neg_b=*/false, b,
        /*c_mod=*/(short)0, acc, /*reuse_a=*/false, /*reuse_b=*/false);
  }
  // C/D layout: row = tm + v + 8*(lane>=16), col = tn + (lane&15)
  float* Yp = Y + (size_t)(tm + ((lane >> 4) << 3)) * N + tn + r;
  for (int v = 0; v < 8; ++v) Yp[(size_t)v * N] = acc[v];
}

// ---------------- elementwise helpers ----------------
__global__ void build_inp_kernel(const float* __restrict__ norm,
                                 const float* __restrict__ pos,
                                 const float* __restrict__ x,
                                 float* __restrict__ inp) {
  int i = blockIdx.x * blockDim.x + threadIdx.x;
  if (i >= NT) return;
  float* o = inp + (size_t)i * 8;
  o[0] = norm[i * 3 + 0]; o[1] = norm[i * 3 + 1]; o[2] = norm[i * 3 + 2];
  o[3] = pos[i * 3 + 0];  o[4] = pos[i * 3 + 1];  o[5] = pos[i * 3 + 2];
  o[6] = x[i * 2 + 0];    o[7] = x[i * 2 + 1];
}

__global__ void bias_relu_kernel(float* __restrict__ acc,
                                 const float* __restrict__ b, int total, int C) {
  int i = blockIdx.x * blockDim.x + threadIdx.x;
  if (i >= total) return;
  float v = acc[i] + b[i % C];
  acc[i] = v > 0.0f ? v : 0.0f;
}

__global__ void build_h1cat_kernel(const float* __restrict__ h1,
                                   const float* __restrict__ inp,
                                   float* __restrict__ h1cat) {
  int i = blockIdx.x * blockDim.x + threadIdx.x;
  if (i >= NT * 72) return;
  int row = i / 72, c = i % 72;
  h1cat[i] = (c < 64) ? h1[(size_t)row * 64 + c] : inp[(size_t)row * 8 + (c - 64)];
}

__global__ void pad_w3_kernel(const float* __restrict__ W3, float* __restrict__ w3p) {
  int i = blockIdx.x * blockDim.x + threadIdx.x;
  if (i >= 140 * 256) return;
  int k = i / 256, n = i % 256;
  w3p[i] = (k < 139) ? W3[(size_t)k * 256 + n] : 0.0f;
}

// ---------------- edge scatter (segment_sum via atomics) ----------------
__global__ void scatter_edges_kernel(const long long* __restrict__ se,
                                     const long long* __restrict__ de,
                                     const float* __restrict__ xw,
                                     float* __restrict__ acc, int E, int C) {
  int i = blockIdx.x * blockDim.x + threadIdx.x;
  int chunks = C >> 2;
  if (i >= E * chunks) return;
  int e = i / chunks;
  int c = (i % chunks) << 2;
  int s = (int)se[e], d = (int)de[e];
  const float* xs = xw + (size_t)s * C + c;
  float* ad = acc + (size_t)d * C + c;
  atomicAdd(ad + 0, xs[0]); atomicAdd(ad + 1, xs[1]);
  atomicAdd(ad + 2, xs[2]); atomicAdd(ad + 3, xs[3]);
}

__global__ void scatter_edges_filtered_kernel(const long long* __restrict__ se,
                                              const long long* __restrict__ de,
                                              const int* __restrict__ mapping,
                                              const float* __restrict__ xw,
                                              float* __restrict__ acc, int E, int C) {
  int i = blockIdx.x * blockDim.x + threadIdx.x;
  int chunks = C >> 2;
  if (i >= E * chunks) return;
  int e = i / chunks;
  int c = (i % chunks) << 2;
  int ns = mapping[(int)se[e]];
  int nd = mapping[(int)de[e]];
  if (ns < 0 || nd < 0) return;
  const float* xs = xw + (size_t)ns * C + c;
  float* ad = acc + (size_t)nd * C + c;
  atomicAdd(ad + 0, xs[0]); atomicAdd(ad + 1, xs[1]);
  atomicAdd(ad + 2, xs[2]); atomicAdd(ad + 3, xs[3]);
}

// ---------------- farthest point sampling ----------------
// one block per graph; 1024 threads x 4 register-resident points; serial S steps
__global__ __launch_bounds__(1024) void fps_kernel(const float* __restrict__ pos,
                                                   int* __restrict__ idx) {
  const int g = blockIdx.x;
  const int t = threadIdx.x;
  const float* pg = pos + (size_t)g * NNODE * 3;
  float px[4], py[4], pz[4], md[4];
  for (int i = 0; i < 4; ++i) {
    int j = t * 4 + i;
    px[i] = pg[j * 3 + 0]; py[i] = pg[j * 3 + 1]; pz[i] = pg[j * 3 + 2];
    md[i] = 3.4e38f;
  }
  __shared__ float scx, scy, scz;
  __shared__ int scur;
  __shared__ float rv[32];
  __shared__ int ri[32];
  if (t == 0) scur = 0;
  __syncthreads();
  for (int s = 0; s < SSAMP; ++s) {
    int cur = scur;
    if (t == 0) idx[g * SSAMP + s] = cur;
    if ((cur >> 2) == t) {
      int l = cur & 3;
      scx = px[l]; scy = py[l]; scz = pz[l];
    }
    __syncthreads();
    float cx = scx, cy = scy, cz = scz;
    float bv = -1.0f; int bi = 0;
    for (int i = 0; i < 4; ++i) {
      float dx = px[i] - cx, dy = py[i] - cy, dz = pz[i] - cz;
      float d = dx * dx + dy * dy + dz * dz;
      md[i] = fminf(md[i], d);
      if (md[i] > bv) { bv = md[i]; bi = t * 4 + i; }
    }
    // wave32 argmax reduce (ties -> lowest index, matching jnp.argmax)
    for (int off = 16; off > 0; off >>= 1) {
      float ov = __shfl_down(bv, off, 32);
      int   oi = __shfl_down(bi, off, 32);
      if (ov > bv || (ov == bv && oi < bi)) { bv = ov; bi = oi; }
    }
    if ((t & 31) == 0) { rv[t >> 5] = bv; ri[t >> 5] = bi; }
    __syncthreads();
    if (t == 0) {
      float best = rv[0]; int besti = ri[0];
      for (int w = 1; w < 32; ++w)
        if (rv[w] > best) { best = rv[w]; besti = ri[w]; }
      scur = besti;
    }
    __syncthreads();
  }
}

// ---------------- radius: first-K-by-index neighbors within R ----------------
// one wave32 per sample; ordered ballot append == reference top_k(score) semantics
__global__ __launch_bounds__(256) void radius_kernel(const float* __restrict__ pos,
                                                     const int* __restrict__ fpsidx,
                                                     int* __restrict__ cols,
                                                     int* __restrict__ nvalid) {
  int wid = (blockIdx.x * blockDim.x + threadIdx.x) >> 5;
  int lane = threadIdx.x & 31;
  if (wid >= GS) return;
  int g = wid / SSAMP;
  const float* pg = pos + (size_t)g * NNODE * 3;
  int qi = fpsidx[wid];
  float qx = pg[qi * 3 + 0], qy = pg[qi * 3 + 1], qz = pg[qi * 3 + 2];
  int cnt = 0;
  for (int c = 0; c < NNODE / 32 && cnt < KNB; ++c) {
    int j = c * 32 + lane;
    float dx = pg[j * 3 + 0] - qx, dy = pg[j * 3 + 1] - qy, dz = pg[j * 3 + 2] - qz;
    bool in = (dx * dx + dy * dy + dz * dz) < R2;
    unsigned long long m = __ballot(in);
    int pre = __popcll(m & ((1ull << lane) - 1ull));
    int slot = cnt + pre;
    if (in && slot < KNB) cols[(size_t)wid * KNB + slot] = j;
    cnt += __popcll(m);
  }
  if (lane == 0) nvalid[wid] = cnt < KNB ? cnt : KNB;
}

// ---------------- PointConv: max over neighbors of [h2_j, pos_j - q_i] ----------------
__global__ __launch_bounds__(160) void pointconv_kernel(const float* __restrict__ h2,
                                                        const float* __restrict__ pos,
                                                        const int* __restrict__ fpsidx,
                                                        const int* __restrict__ cols,
                                                        const int* __restrict__ nvalid,
                                                        float* __restrict__ xint) {
  int sid = blockIdx.x;
  int d = threadIdx.x;
  if (d >= 131) return;
  int g = sid / SSAMP;
  const float* pg = pos + (size_t)g * NNODE * 3;
  const float* hg = h2 + (size_t)g * NNODE * 128;
  int qi = fpsidx[sid];
  float qc = (d >= 128) ? pg[qi * 3 + (d - 128)] : 0.0f;
  int nv = nvalid[sid];
  const int* cp = cols + (size_t)sid * KNB;
  float best = -INFINITY;
  for (int k = 0; k < nv; ++k) {
    int col = cp[k];
    float v = (d < 128) ? hg[(size_t)col * 128 + d] : (pg[col * 3 + (d - 128)] - qc);
    best = fmaxf(best, v);
  }
  xint[(size_t)sid * 131 + d] = best;
}

// ---------------- filter_adj mapping ----------------
__global__ void mapping_set_kernel(const int* __restrict__ fpsidx, int* __restrict__ mapping) {
  int i = blockIdx.x * blockDim.x + threadIdx.x;
  if (i >= GS) return;
  int g = i / SSAMP;
  mapping[g * NNODE + fpsidx[i]] = i;
}

__global__ void build_h3in_kernel(const float* __restrict__ xint,
                                  const float* __restrict__ inp,
                                  const int* __restrict__ fpsidx,
                                  float* __restrict__ h3in) {
  int i = blockIdx.x * blockDim.x + threadIdx.x;
  if (i >= GS * 140) return;
  int row = i / 140, c = i % 140;
  float v;
  if (c < 131) v = xint[(size_t)row * 131 + c];
  else if (c < 139) {
    int g = row / SSAMP;
    v = inp[(size_t)(g * NNODE + fpsidx[row]) * 8 + (c - 131)];
  } else v = 0.0f;   // K padding column
  h3in[i] = v;
}

// ---------------- global max pool + classifier head ----------------
__global__ void pool_kernel(const float* __restrict__ h3, float* __restrict__ pooled) {
  int g = blockIdx.x, c = threadIdx.x;   // 256 threads
  float m = -INFINITY;
  const float* p = h3 + (size_t)g * SSAMP * 256 + c;
  for (int s = 0; s < SSAMP; ++s) m = fmaxf(m, p[(size_t)s * 256]);
  pooled[g * 256 + c] = m;
}

__global__ __launch_bounds__(128) void head_kernel(const float* __restrict__ pooled,
                                                   const float* __restrict__ lw,
                                                   const float* __restrict__ lb,
                                                   float* __restrict__ out) {
  __shared__ float lg[GNUM][CCLS];
  int t = threadIdx.x;
  if (t < GNUM * CCLS) {
    int g = t / CCLS, c = t % CCLS;
    float s = lb[c];
    for (int k = 0; k < 256; ++k) s += pooled[g * 256 + k] * lw[k * CCLS + c];
    lg[g][c] = s;
  }
  __syncthreads();
  if (t < GNUM) {
    float mx = -INFINITY;
    for (int c = 0; c < CCLS; ++c) mx = fmaxf(mx, lg[t][c]);
    float se = 0.0f;
    for (int c = 0; c < CCLS; ++c) se += expf(lg[t][c] - mx);
    float ls = logf(se);
    for (int c = 0; c < CCLS; ++c) {
      float o = lg[t][c] - mx - ls;
      out[t * CCLS + c] = o;                      // log_softmax
      out[GNUM * CCLS + t * CCLS + c] = expf(o);  // softmax(log_softmax) == softmax
    }
  }
}

// ---------------- driver ----------------
extern "C" void kernel_launch(void* const* d_in, const int* in_sizes, int n_in,
                              void* d_out, int out_size, void* d_ws, size_t ws_size,
                              hipStream_t stream) {
  const float* norm = (const float*)d_in[0];
  const float* pos  = (const float*)d_in[1];
  const float* x    = (const float*)d_in[2];
  const long long* ei = (const long long*)d_in[3];   // int64 [2, E]
  // d_in[4] = batch (unused; derived arithmetically)
  const float* W1 = (const float*)d_in[5];
  const float* b1 = (const float*)d_in[6];
  const float* W2 = (const float*)d_in[7];
  const float* b2 = (const float*)d_in[8];
  const float* W3 = (const float*)d_in[9];
  const float* b3 = (const float*)d_in[10];
  const float* lw = (const float*)d_in[11];
  const float* lb = (const float*)d_in[12];
  float* out = (float*)d_out;

  const long long* esrc = ei;
  const long long* edst = ei + ETOT;

  // workspace carve-up (all sizes are multiples of 256 B)
  char* ws = (char*)d_ws;
  size_t o = 0;
  auto take = [&](size_t bytes) { char* p = ws + o; o += bytes; return p; };
  float* inp    = (float*)take((size_t)NT * 8 * 4);
  float* xw1    = (float*)take((size_t)NT * 64 * 4);
  float* acc1   = (float*)take((size_t)NT * 64 * 4);
  float* h1cat  = (float*)take((size_t)NT * 72 * 4);
  float* xw2    = (float*)take((size_t)NT * 128 * 4);
  float* acc2   = (float*)take((size_t)NT * 128 * 4);
  int*   fpsidx = (int*)take((size_t)GS * 4);
  int*   cols   = (int*)take((size_t)GS * KNB * 4);
  int*   nvalid = (int*)take((size_t)GS * 4);
  float* xint   = (float*)take((size_t)GS * 131 * 4);
  int*   mapping= (int*)take((size_t)NT * 4);
  float* h3in   = (float*)take((size_t)GS * 140 * 4);
  float* w3p    = (float*)take((size_t)140 * 256 * 4);
  float* xw3    = (float*)take((size_t)GS * 256 * 4);
  float* acc3   = (float*)take((size_t)GS * 256 * 4);
  float* pooled = (float*)take((size_t)GNUM * 256 * 4);
  (void)ws_size; (void)n_in; (void)in_sizes; (void)out_size;

  // 1) input concat [NT,8]
  build_inp_kernel<<<(NT + 255) / 256, 256, 0, stream>>>(norm, pos, x, inp);

  // 2) layer 1: xw1 = inp @ W1  (WMMA), scatter, bias+relu
  gemm_wmma_f32<<<((NT / 16) * (64 / 16)) / 4, 128, 0, stream>>>(inp, W1, xw1, NT, 8, 64);
  hipMemsetAsync(acc1, 0, (size_t)NT * 64 * 4, stream);
  scatter_edges_kernel<<<(ETOT * 16 + 255) / 256, 256, 0, stream>>>(esrc, edst, xw1, acc1, ETOT, 64);
  bias_relu_kernel<<<(NT * 64 + 255) / 256, 256, 0, stream>>>(acc1, b1, NT * 64, 64);

  // 3) layer 2: h1cat = [h1, inp]; xw2 = h1cat @ W2 (WMMA); scatter; bias+relu
  build_h1cat_kernel<<<(NT * 72 + 255) / 256, 256, 0, stream>>>(acc1, inp, h1cat);
  gemm_wmma_f32<<<((NT / 16) * (128 / 16)) / 4, 128, 0, stream>>>(h1cat, W2, xw2, NT, 72, 128);
  hipMemsetAsync(acc2, 0, (size_t)NT * 128 * 4, stream);
  scatter_edges_kernel<<<(ETOT * 32 + 255) / 256, 256, 0, stream>>>(esrc, edst, xw2, acc2, ETOT, 128);
  bias_relu_kernel<<<(NT * 128 + 255) / 256, 256, 0, stream>>>(acc2, b2, NT * 128, 128);

  // 4) FPS (serial per graph, register-resident points), radius, PointConv
  fps_kernel<<<GNUM, 1024, 0, stream>>>(pos, fpsidx);
  radius_kernel<<<(GS * 32 + 255) / 256, 256, 0, stream>>>(pos, fpsidx, cols, nvalid);
  pointconv_kernel<<<GS, 160, 0, stream>>>(acc2, pos, fpsidx, cols, nvalid, xint);

  // 5) filter_adj mapping
  hipMemsetAsync(mapping, 0xFF, (size_t)NT * 4, stream);   // -1
  mapping_set_kernel<<<(GS + 255) / 256, 256, 0, stream>>>(fpsidx, mapping);

  // 6) layer 3: h3in = [x_int, inp[idx]] (K padded 139->140); WMMA GEMM; filtered scatter; bias+relu
  build_h3in_kernel<<<(GS * 140 + 255) / 256, 256, 0, stream>>>(xint, inp, fpsidx, h3in);
  pad_w3_kernel<<<(140 * 256 + 255) / 256, 256, 0, stream>>>(W3, w3p);
  gemm_wmma_f32<<<((GS / 16) * (256 / 16)) / 4, 128, 0, stream>>>(h3in, w3p, xw3, GS, 140, 256);
  hipMemsetAsync(acc3, 0, (size_t)GS * 256 * 4, stream);
  scatter_edges_filtered_kernel<<<(ETOT * 64 + 255) / 256, 256, 0, stream>>>(esrc, edst, mapping, xw3, acc3, ETOT, 256);
  bias_relu_kernel<<<(GS * 256 + 255) / 256, 256, 0, stream>>>(acc3, b3, GS * 256, 256);

  // 7) global max pool + classifier + log_softmax/softmax
  pool_kernel<<<GNUM, 256, 0, stream>>>(acc3, pooled);
  head_kernel<<<1, 128, 0, stream>>>(pooled, lw, lb, out);
}